// BEVConvS_19696720019810
// MI455X (gfx1250) — compile-verified
//
#include <hip/hip_runtime.h>
#include <hip/hip_bf16.h>

// ---------------------------------------------------------------------------
// BEVConvS pipeline for MI455X (gfx1250, wave32).
//   k_init_bev       : bev_z = -10, bev_i = 0                [4,2,1024,1024]
//   k_scatter        : scatter-max points into BEV
//   k_conv1_pool     : conv 2->8 3x3 + BN + ReLU + 2x2 pool  -> [4,8,512,512]
//   k_conv2_pool     : grouped conv 8->16 + BN + ReLU + pool -> [4,16,256,256]
//   k_dwpw           : depthwise 16 + shuffle(id) + pw 16->32 + BN + ReLU
//   k_transpose_w3   : w3[co][ci][t] -> wT[t*32+ci][co]  (tap-major K, 288x64)
//   k_conv3_wmma     : TENSOR_LOAD_TO_LDS staging + V_WMMA_F32_16X16X4_F32
//   k_pool3          : 2x2 max pool -> d_out [4,64,128,128]
// ---------------------------------------------------------------------------

typedef float v2f __attribute__((ext_vector_type(2)));
typedef float v8f __attribute__((ext_vector_type(8)));
typedef unsigned int u32x4 __attribute__((ext_vector_type(4)));
typedef int i32x8 __attribute__((ext_vector_type(8)));
typedef int i32x4 __attribute__((ext_vector_type(4)));

#define BATCH 4
#define HH 1024
#define WW 1024
#define BN_EPS 1e-5f

__device__ __forceinline__ void atomicMaxF(float* addr, float val) {
    int* ia = (int*)addr;
    int old = __float_as_int(*addr);
    while (__int_as_float(old) < val) {
        int assumed = old;
        old = atomicCAS(ia, assumed, __float_as_int(val));
        if (old == assumed) break;
    }
}

// ------------------------------ rasterize ----------------------------------
__global__ void k_init_bev(float* __restrict__ bev) {
    int idx = blockIdx.x * blockDim.x + threadIdx.x;
    const int total = BATCH * 2 * HH * WW;
    if (idx >= total) return;
    int plane = (idx >> 20) & 1;              // [b][c][y][x], c at bit 20
    bev[idx] = (plane == 0) ? -10.0f : 0.0f;
}

__global__ void k_scatter(const float* __restrict__ pts, int n,
                          float* __restrict__ bev) {
    int i = blockIdx.x * blockDim.x + threadIdx.x;
    if (i >= n) return;
    const float xs = 1024.0f / 69.12f;
    const float ys = 1024.0f / 79.36f;
    float bi = pts[i * 5 + 0];
    float px = pts[i * 5 + 1];
    float py = pts[i * 5 + 2];
    float pz = pts[i * 5 + 3];
    float in = pts[i * 5 + 4];
    int xp = (int)(px * xs);                      // trunc toward zero
    int yp = (int)((py + 39.68f) * ys);
    if (xp < 0 || xp >= WW || yp < 0 || yp >= HH) return;
    int b = (int)bi;
    float* z = bev + (((size_t)(b * 2 + 0) * HH + yp) * WW + xp);
    float* s = bev + (((size_t)(b * 2 + 1) * HH + yp) * WW + xp);
    atomicMaxF(z, pz);
    atomicMaxF(s, in);
}

// -------------------- conv1 (2->8) + BN + ReLU + pool ----------------------
__global__ void k_conv1_pool(const float* __restrict__ bev,
                             const float* __restrict__ w0, const float* __restrict__ b0,
                             const float* __restrict__ g0, const float* __restrict__ be0,
                             const float* __restrict__ m0, const float* __restrict__ v0,
                             float* __restrict__ x1) {
    int idx = blockIdx.x * blockDim.x + threadIdx.x;
    const int total = BATCH * 8 * 512 * 512;
    if (idx >= total) return;
    int px = idx & 511; int t = idx >> 9;
    int py = t & 511;   t >>= 9;
    int co = t & 7;     int b = t >> 3;
    float s   = g0[co] * rsqrtf(v0[co] + BN_EPS);
    float off = be0[co] - m0[co] * s;
    float bias = b0[co];
    float mx = -3.402823466e38f;
    #pragma unroll
    for (int i = 0; i < 2; ++i)
    #pragma unroll
    for (int j = 0; j < 2; ++j) {
        int oy = 2 * py + i, ox = 2 * px + j;
        float acc = bias;
        #pragma unroll
        for (int ci = 0; ci < 2; ++ci)
        #pragma unroll
        for (int ky = 0; ky < 3; ++ky) {
            int y = oy + ky - 1;
            if (y < 0 || y >= HH) continue;
            #pragma unroll
            for (int kx = 0; kx < 3; ++kx) {
                int x = ox + kx - 1;
                if (x < 0 || x >= WW) continue;
                acc += bev[((size_t)(b * 2 + ci) * HH + y) * WW + x] *
                       w0[((co * 2 + ci) * 3 + ky) * 3 + kx];
            }
        }
        mx = fmaxf(mx, fmaxf(acc * s + off, 0.0f));
    }
    x1[idx] = mx;
}

// ------------- conv2 grouped (8->16, groups=8) + BN + ReLU + pool ----------
__global__ void k_conv2_pool(const float* __restrict__ x1,
                             const float* __restrict__ w1, const float* __restrict__ b1,
                             const float* __restrict__ g1, const float* __restrict__ be1,
                             const float* __restrict__ m1, const float* __restrict__ v1,
                             float* __restrict__ x2) {
    int idx = blockIdx.x * blockDim.x + threadIdx.x;
    const int total = BATCH * 16 * 256 * 256;
    if (idx >= total) return;
    int px = idx & 255; int t = idx >> 8;
    int py = t & 255;   t >>= 8;
    int co = t & 15;    int b = t >> 4;
    int ci = co >> 1;                                  // group = co/2, 1 in-ch/group
    float s   = g1[co] * rsqrtf(v1[co] + BN_EPS);
    float off = be1[co] - m1[co] * s;
    float bias = b1[co];
    float mx = -3.402823466e38f;
    #pragma unroll
    for (int i = 0; i < 2; ++i)
    #pragma unroll
    for (int j = 0; j < 2; ++j) {
        int oy = 2 * py + i, ox = 2 * px + j;
        float acc = bias;
        #pragma unroll
        for (int ky = 0; ky < 3; ++ky) {
            int y = oy + ky - 1;
            if (y < 0 || y >= 512) continue;
            #pragma unroll
            for (int kx = 0; kx < 3; ++kx) {
                int x = ox + kx - 1;
                if (x < 0 || x >= 512) continue;
                acc += x1[((size_t)(b * 8 + ci) * 512 + y) * 512 + x] *
                       w1[(co * 3 + ky) * 3 + kx];
            }
        }
        mx = fmaxf(mx, fmaxf(acc * s + off, 0.0f));
    }
    x2[idx] = mx;
}

// ---------- depthwise(16) + shuffle(identity) + pointwise 16->32 -----------
__global__ void k_dwpw(const float* __restrict__ x2,
                       const float* __restrict__ wdw, const float* __restrict__ bdw,
                       const float* __restrict__ wpw, const float* __restrict__ bpw,
                       const float* __restrict__ g2, const float* __restrict__ be2,
                       const float* __restrict__ m2, const float* __restrict__ v2,
                       float* __restrict__ xpw) {
    int idx = blockIdx.x * blockDim.x + threadIdx.x;
    const int total = BATCH * 256 * 256;
    if (idx >= total) return;
    int x = idx & 255; int t = idx >> 8;
    int y = t & 255;   int b = t >> 8;
    float dw[16];
    #pragma unroll
    for (int c = 0; c < 16; ++c) {
        float acc = bdw[c];
        #pragma unroll
        for (int ky = 0; ky < 3; ++ky) {
            int yy = y + ky - 1;
            if (yy < 0 || yy >= 256) continue;
            #pragma unroll
            for (int kx = 0; kx < 3; ++kx) {
                int xx = x + kx - 1;
                if (xx < 0 || xx >= 256) continue;
                acc += x2[((size_t)(b * 16 + c) * 256 + yy) * 256 + xx] *
                       wdw[(c * 3 + ky) * 3 + kx];
            }
        }
        dw[c] = acc;
    }
    for (int co = 0; co < 32; ++co) {
        float acc = bpw[co];
        #pragma unroll
        for (int c = 0; c < 16; ++c) acc += dw[c] * wpw[co * 16 + c];
        float s = g2[co] * rsqrtf(v2[co] + BN_EPS);
        acc = fmaxf(acc * s + (be2[co] - m2[co] * s), 0.0f);
        xpw[((size_t)(b * 32 + co) * 256 + y) * 256 + x] = acc;
    }
}

// --------------------- weight transpose for conv3 B ------------------------
// Tap-major K ordering: k = t*32 + ci  (t = ky*3+kx in 0..8, ci in 0..31)
__global__ void k_transpose_w3(const float* __restrict__ w3, float* __restrict__ wT) {
    int i = blockIdx.x * blockDim.x + threadIdx.x;       // i = k*64 + co
    if (i >= 288 * 64) return;
    int k = i >> 6, co = i & 63;
    int t = k >> 5, ci = k & 31;
    wT[i] = w3[co * 288 + ci * 9 + t];
}

// ------------- conv3 (32->64, 3x3) via WMMA implicit GEMM ------------------
// Block: 128 threads = 4 waves. Each block: one output row (oy), 64 pixels.
// Activation halo tile staged via TENSOR_LOAD_TO_LDS (TDM), K tap-major,
// each wave: 16 pixels (M) x 64 out-channels (4 N-tiles), 288 wmma/wave.
__global__ __launch_bounds__(128) void k_conv3_wmma(
        const float* __restrict__ xin,   // [B,32,256,256]
        const float* __restrict__ wT,    // [288,64] tap-major
        const float* __restrict__ b3, const float* __restrict__ g3,
        const float* __restrict__ be3, const float* __restrict__ m3,
        const float* __restrict__ v3,
        float* __restrict__ xout) {      // [B,64,256,256]
    __shared__ float Al[32 * 3 * 66];    // halo tile: [ci][row 0..2][col 0..65]

    int blk = blockIdx.x;                // 4 tiles/row * 256 rows * 4 batches
    int row = blk >> 2;                  // b*256 + oy
    int b   = row >> 8;
    int oy  = row & 255;
    int ox0 = (blk & 3) * 64;
    int tid = threadIdx.x;

    // ---- TDM stage: 3-D tile x=66 (cols), y=3 (rows), z=32 (channels) -----
    // Strided copy from tile start (may hang off image edges; those reads
    // wrap within the workspace allocation and are zero-fixed below).
    if (tid < 32) {                      // wave 0 issues the tensor DMA
        const float* gbase = xin + (((size_t)(b * 32) * 256 + (oy - 1)) * 256 + (ox0 - 1));
        unsigned long long ga = (unsigned long long)(uintptr_t)gbase;
        unsigned lds0 = (unsigned)(uintptr_t)(void*)&Al[0];
        u32x4 g0;
        g0[0] = 1u;                                            // count=1 (valid, user)
        g0[1] = lds0;                                          // lds_addr (bytes)
        g0[2] = (unsigned)(ga & 0xffffffffu);                  // global_addr[31:0]
        g0[3] = (unsigned)((ga >> 32) & 0x1ffffffu) | (2u << 30); // addr[56:32] | type=2
        i32x8 g1;
        g1[0] = (2 << 16);                 // workgroup_mask=0, data_size=2 (4B)
        g1[1] = (int)((0x100000u & 0xffffu) << 16);            // tensor_dim0 lo16 (big)
        g1[2] = (int)((0x100000u >> 16) | ((0x100000u & 0xffffu) << 16)); // d0 hi | d1 lo
        g1[3] = (int)((0x100000u >> 16) | (66u << 16));        // d1 hi | tile_dim0=66
        g1[4] = (int)(3u | (32u << 16));   // tile_dim1=3 | tile_dim2=32
        g1[5] = 256;                       // tensor_dim0_stride (row pitch)
        g1[6] = (int)((65536u & 0xffffu) << 16);               // stride0 hi | stride1 lo16
        g1[7] = (int)(65536u >> 16);       // tensor_dim1_stride hi (channel pitch 65536)
        i32x4 g2;
        g2[0] = 0x100000;                  // tensor_dim2 (big)
        g2[1] = 0; g2[2] = 0; g2[3] = 0;   // dim3 / stride2 / tile_dim3 unused
        i32x4 g3d; g3d[0] = 0; g3d[1] = 0; g3d[2] = 0; g3d[3] = 0;
        i32x8 g4; g4[0] = 0; g4[1] = 0; g4[2] = 0; g4[3] = 0;
        g4[4] = 0; g4[5] = 0; g4[6] = 0; g4[7] = 0;
        __builtin_amdgcn_tensor_load_to_lds(g0, g1, g2, g3d, g4, 0);
        __builtin_amdgcn_s_wait_tensorcnt(0);
    }
    __syncthreads();

    // ---- zero-fix halo entries that fell outside the image ----------------
    if (oy == 0)
        for (int i = tid; i < 32 * 66; i += 128) Al[(i / 66) * 198 + (i % 66)] = 0.0f;
    if (oy == 255)
        for (int i = tid; i < 32 * 66; i += 128) Al[(i / 66) * 198 + 132 + (i % 66)] = 0.0f;
    if (ox0 == 0)
        for (int i = tid; i < 32 * 3; i += 128) Al[(i / 3) * 198 + (i % 3) * 66] = 0.0f;
    if (ox0 == 192)
        for (int i = tid; i < 32 * 3; i += 128) Al[(i / 3) * 198 + (i % 3) * 66 + 65] = 0.0f;
    __syncthreads();

    int wave = tid >> 5;
    int lane = tid & 31;
    int nloc = lane & 15;                // A: pixel row M; B/D: local column N
    int hi   = lane >> 4;                // K sub-pair selector for A/B fragments
    int mcol = wave * 16 + nloc;         // this lane's A pixel offset within tile

    v8f acc0 = {}, acc1 = {}, acc2 = {}, acc3 = {};

    // K loop, tap-major: k = t*32 + ci; per-step A rows K=ka,ka+1 per lane.
    #pragma unroll
    for (int t = 0; t < 9; ++t) {
        const float* Ap = Al + (t / 3) * 66 + mcol + (t % 3);  // + ci*198
        const float* Wt = wT + t * 32 * 64;
        for (int c0 = 0; c0 < 32; c0 += 4) {
            int cl = c0 + 2 * hi;
            v2f a;
            a.x = Ap[cl * 198];
            a.y = Ap[cl * 198 + 198];
            const float* w0p = Wt + (size_t)cl * 64;
            const float* w1p = w0p + 64;
            v2f bq;
            bq.x = w0p[nloc];      bq.y = w1p[nloc];
            acc0 = __builtin_amdgcn_wmma_f32_16x16x4_f32(false, a, false, bq, (short)0, acc0, false, false);
            bq.x = w0p[16 + nloc]; bq.y = w1p[16 + nloc];
            acc1 = __builtin_amdgcn_wmma_f32_16x16x4_f32(false, a, false, bq, (short)0, acc1, false, false);
            bq.x = w0p[32 + nloc]; bq.y = w1p[32 + nloc];
            acc2 = __builtin_amdgcn_wmma_f32_16x16x4_f32(false, a, false, bq, (short)0, acc2, false, false);
            bq.x = w0p[48 + nloc]; bq.y = w1p[48 + nloc];
            acc3 = __builtin_amdgcn_wmma_f32_16x16x4_f32(false, a, false, bq, (short)0, acc3, false, false);
        }
    }

    // Epilogue: D layout -> lane L: N = L%16, VGPR r: M = r + 8*(L/16).
    #pragma unroll
    for (int nt = 0; nt < 4; ++nt) {
        v8f acc = (nt == 0) ? acc0 : (nt == 1) ? acc1 : (nt == 2) ? acc2 : acc3;
        int co = nt * 16 + nloc;
        float s   = g3[co] * rsqrtf(v3[co] + BN_EPS);
        float off = be3[co] - m3[co] * s + b3[co] * s;   // (conv + bias)*s + (be - m*s)
        float* outp = xout + ((size_t)(b * 64 + co) * 256 + oy) * 256;
        #pragma unroll
        for (int r = 0; r < 8; ++r) {
            int pxr = ox0 + wave * 16 + r + 8 * hi;
            outp[pxr] = fmaxf(acc[r] * s + off, 0.0f);
        }
    }
}

// ------------------------------ final pool ---------------------------------
__global__ void k_pool3(const float* __restrict__ x3, float* __restrict__ out) {
    int idx = blockIdx.x * blockDim.x + threadIdx.x;
    const int total = BATCH * 64 * 128 * 128;
    if (idx >= total) return;
    int px = idx & 127; int t = idx >> 7;
    int py = t & 127;   t >>= 7;
    int c  = t & 63;    int b = t >> 6;
    const float* p = x3 + ((size_t)(b * 64 + c) * 256 + 2 * py) * 256 + 2 * px;
    float m = fmaxf(fmaxf(p[0], p[1]), fmaxf(p[256], p[257]));
    out[idx] = m;
}

// ---------------------------------------------------------------------------
extern "C" void kernel_launch(void* const* d_in, const int* in_sizes, int n_in,
                              void* d_out, int out_size, void* d_ws, size_t ws_size,
                              hipStream_t stream) {
    const float* points = (const float*)d_in[0];
    int npts = in_sizes[0] / 5;
    const float* w0 = (const float*)d_in[2];
    const float* b0 = (const float*)d_in[3];
    const float* g0 = (const float*)d_in[4];
    const float* be0 = (const float*)d_in[5];
    const float* m0 = (const float*)d_in[6];
    const float* v0 = (const float*)d_in[7];
    const float* w1 = (const float*)d_in[8];
    const float* b1 = (const float*)d_in[9];
    const float* g1 = (const float*)d_in[10];
    const float* be1 = (const float*)d_in[11];
    const float* m1 = (const float*)d_in[12];
    const float* v1 = (const float*)d_in[13];
    const float* wdw = (const float*)d_in[14];
    const float* bdw = (const float*)d_in[15];
    const float* wpw = (const float*)d_in[16];
    const float* bpw = (const float*)d_in[17];
    const float* g2 = (const float*)d_in[18];
    const float* be2 = (const float*)d_in[19];
    const float* m2 = (const float*)d_in[20];
    const float* v2 = (const float*)d_in[21];
    const float* w3 = (const float*)d_in[22];
    const float* b3 = (const float*)d_in[23];
    const float* g3 = (const float*)d_in[24];
    const float* be3 = (const float*)d_in[25];
    const float* m3 = (const float*)d_in[26];
    const float* v3 = (const float*)d_in[27];

    float* ws = (float*)d_ws;
    // Workspace layout (float offsets) with liveness-based reuse:
    float* BEV = ws + 0;                      //  8,388,608  [4,2,1024,1024]
    float* X1  = ws + 8388608;                //  8,388,608  [4,8,512,512]
    float* X2  = ws + 0;                      //  4,194,304  [4,16,256,256]  (reuses BEV)
    float* XPW = ws + 4194304;                //  8,388,608  [4,32,256,256]  (reuses BEV tail + X1 head)
    float* X3  = ws + 16777216;               // 16,777,216  [4,64,256,256]
    float* WT  = ws + 33554432;               //     18,432  [288,64]

    const int TB = 256;
    k_init_bev<<<(BATCH * 2 * HH * WW + TB - 1) / TB, TB, 0, stream>>>(BEV);
    k_scatter<<<(npts + TB - 1) / TB, TB, 0, stream>>>(points, npts, BEV);
    k_conv1_pool<<<(BATCH * 8 * 512 * 512 + TB - 1) / TB, TB, 0, stream>>>(
        BEV, w0, b0, g0, be0, m0, v0, X1);
    k_conv2_pool<<<(BATCH * 16 * 256 * 256 + TB - 1) / TB, TB, 0, stream>>>(
        X1, w1, b1, g1, be1, m1, v1, X2);
    k_dwpw<<<(BATCH * 256 * 256 + TB - 1) / TB, TB, 0, stream>>>(
        X2, wdw, bdw, wpw, bpw, g2, be2, m2, v2, XPW);
    k_transpose_w3<<<(288 * 64 + TB - 1) / TB, TB, 0, stream>>>(w3, WT);
    k_conv3_wmma<<<BATCH * 256 * 4, 128, 0, stream>>>(
        XPW, WT, b3, g3, be3, m3, v3, X3);
    k_pool3<<<(BATCH * 64 * 128 * 128 + TB - 1) / TB, TB, 0, stream>>>(
        X3, (float*)d_out);
}